// BeamTreeEnsemble_28200755265904
// MI455X (gfx1250) — compile-verified
//
#include <hip/hip_runtime.h>

// ---------------- problem constants (from reference) ----------------
#define NUM_TREES       256
#define MAX_TREE_DEPTH  10
#define NUM_NODES       2047        // complete binary tree, depth 10
#define N_FEATURES      256
#define N_CLASSES       8
#define BATCH           32768

// ---------------- tiling ----------------
#define TILE_SAMPLES    256         // samples per block (== blockDim.x)
#define TREES_PER_BLOCK 8           // trees amortized per x-tile
#define XROW            260         // padded LDS row stride in floats (TDM pad: +4 DW / 256 DW)
#define NODE_SLOTS      8           // ceil(NUM_NODES / TILE_SAMPLES) register-staging slots

// LDS budget: x tile 256*260*4 = 266240 B + tree table 2048*16 = 32768 B = 299008 B (< 320 KB/WGP)
#define SMEM_BYTES      ((size_t)TILE_SAMPLES * XROW * 4 + 2048 * 16)

typedef __attribute__((ext_vector_type(4))) unsigned int u32x4;
typedef __attribute__((ext_vector_type(4))) int          i32x4;
typedef __attribute__((ext_vector_type(8))) int          i32x8;
typedef __attribute__((ext_vector_type(4))) float        f32x4;   // native vector: OK for NT builtins

#if defined(__HIP_DEVICE_COMPILE__) && __has_builtin(__builtin_amdgcn_tensor_load_to_lds)
#define HAVE_TDM 1
#else
#define HAVE_TDM 0
#endif

__global__ __launch_bounds__(TILE_SAMPLES)
void tree_ensemble_kernel(const float* __restrict__ x,
                          const float* __restrict__ lefts,
                          const float* __restrict__ rights,
                          const int*   __restrict__ features,
                          const float* __restrict__ thresholds,
                          const float* __restrict__ values,
                          float*       __restrict__ out)
{
    extern __shared__ __align__(16) unsigned char smem[];
    float* x_lds    = (float*)smem;                                      // [256][260] padded
    f32x4* tree_lds = (f32x4*)(smem + (size_t)TILE_SAMPLES * XROW * 4);  // [2048] AoS nodes

    const int t       = threadIdx.x;
    const int sample0 = blockIdx.x * TILE_SAMPLES;
    const int tree0   = blockIdx.y * TREES_PER_BLOCK;

    // ---- 1) issue TDM for the x tile [256 x 256] f32 (no wait yet: overlap tree-0 staging) ----
#if HAVE_TDM
    if (t < 32) {   // wave 0 issues the DMA once
        unsigned           lds_addr = (unsigned)(unsigned long long)(void*)x_lds;
        unsigned long long ga       = (unsigned long long)(const void*)
                                      (x + (size_t)sample0 * N_FEATURES);
        // D# group 0: count=1 | lds_addr | global_addr[56:0] | type=2
        u32x4 g0 = { 1u,
                     lds_addr,
                     (unsigned)(ga & 0xFFFFFFFFu),
                     (unsigned)((ga >> 32) & 0x1FFFFFFu) | (2u << 30) };
        // group 1: data_size=4B | pad_enable | pad_interval=256DW | pad_amount=4DW;
        //          tensor_dim0=256, tensor_dim1=32768, tile=256x256, dim0_stride=256
        i32x8 g1 = { (int)((2u << 16) | (1u << 20) | (7u << 22) | (3u << 25)),
                     (int)(256u << 16),
                     (int)(0x8000u << 16),
                     (int)(256u << 16),
                     256,
                     256,
                     0, 0 };
        i32x4 gz = { 0, 0, 0, 0 };
#if defined(__clang_major__) && (__clang_major__ >= 23)
        i32x8 gz8 = { 0, 0, 0, 0, 0, 0, 0, 0 };
        __builtin_amdgcn_tensor_load_to_lds(g0, g1, gz, gz, gz8, 0);
#else
        __builtin_amdgcn_tensor_load_to_lds(g0, g1, gz, gz, 0);
#endif
    }
#else
    for (int i = t; i < TILE_SAMPLES * (N_FEATURES / 4); i += TILE_SAMPLES) {
        int s = i >> 6, c = i & 63;
        f32x4 v = ((const f32x4*)(x + (size_t)(sample0 + s) * N_FEATURES))[c];
        *(f32x4*)&x_lds[s * XROW + c * 4] = v;
    }
#endif

    // ---- 2) stage tree 0 into registers (overlaps the TDM DMA), commit to LDS ----
    f32x4 stage[NODE_SLOTS];
    {
        const size_t nb = (size_t)tree0 * NUM_NODES;
#pragma unroll
        for (int i = 0; i < NODE_SLOTS; ++i) {
            int k = t + i * TILE_SAMPLES;
            if (k < NUM_NODES) {
                f32x4 n;
                n.x = __int_as_float(features[nb + k]);
                n.y = thresholds[nb + k];
                n.z = lefts[nb + k];
                n.w = rights[nb + k];
                stage[i] = n;
            }
        }
#pragma unroll
        for (int i = 0; i < NODE_SLOTS; ++i) {
            int k = t + i * TILE_SAMPLES;
            if (k < NUM_NODES) tree_lds[k] = stage[i];
        }
    }
#if HAVE_TDM
    __builtin_amdgcn_s_wait_tensorcnt(0);   // only the issuing wave has TENSORcnt>0; no-op elsewhere
#endif
    __syncthreads();                        // x tile + tree-0 table visible to all waves

    const float* xrow = &x_lds[(unsigned)t * XROW];

    // ---- 3) per-tree: prefetch next tree into registers, traverse current from LDS ----
    for (int j = 0; j < TREES_PER_BLOCK; ++j) {
        const int    tree = tree0 + j;
        const size_t nb   = (size_t)tree * NUM_NODES;

        // issue next tree's global loads now; their latency hides under the
        // 10-step dependent LDS traversal chain below
        if (j + 1 < TREES_PER_BLOCK) {
            const size_t nb2 = nb + NUM_NODES;
#pragma unroll
            for (int i = 0; i < NODE_SLOTS; ++i) {
                int k = t + i * TILE_SAMPLES;
                if (k < NUM_NODES) {
                    f32x4 n;
                    n.x = __int_as_float(features[nb2 + k]);
                    n.y = thresholds[nb2 + k];
                    n.z = lefts[nb2 + k];
                    n.w = rights[nb2 + k];
                    stage[i] = n;
                }
            }
        }

        int idx = 0;
#pragma unroll
        for (int d = 0; d < MAX_TREE_DEPTH; ++d) {
            f32x4 n    = tree_lds[idx];               // ds_load_b128 gather
            int   feat = __float_as_int(n.x);
            float fv   = xrow[feat];                  // ds_load_b32, bank = (4t+feat)%64
            idx = (int)(fv >= n.y ? n.w : n.z);
        }

        // leaf values: 32 contiguous bytes from the L2-resident table
        const f32x4* vp = (const f32x4*)(values + (nb + (size_t)idx) * N_CLASSES);
        f32x4 v0 = vp[0];
        f32x4 v1 = vp[1];
        f32x4* op = (f32x4*)(out +
                    (((size_t)(sample0 + t) * NUM_TREES) + tree) * N_CLASSES);
        // write-once 256 MB stream: non-temporal so it doesn't evict the reused
        // x/tree/values working set from the 192 MB L2
        __builtin_nontemporal_store(v0, op);
        __builtin_nontemporal_store(v1, op + 1);

        if (j + 1 < TREES_PER_BLOCK) {
            __syncthreads();                          // all readers of tree_lds done
#pragma unroll
            for (int i = 0; i < NODE_SLOTS; ++i) {
                int k = t + i * TILE_SAMPLES;
                if (k < NUM_NODES) tree_lds[k] = stage[i];
            }
            __syncthreads();                          // next table visible
        }
    }
}

extern "C" void kernel_launch(void* const* d_in, const int* in_sizes, int n_in,
                              void* d_out, int out_size, void* d_ws, size_t ws_size,
                              hipStream_t stream) {
    (void)in_sizes; (void)n_in; (void)out_size; (void)d_ws; (void)ws_size;
    // setup_inputs order: x, lefts, rights, features, thresholds, values, nodes_offset
    const float* x          = (const float*)d_in[0];
    const float* lefts      = (const float*)d_in[1];
    const float* rights     = (const float*)d_in[2];
    const int*   features   = (const int*)  d_in[3];
    const float* thresholds = (const float*)d_in[4];
    const float* values     = (const float*)d_in[5];
    float*       out        = (float*)d_out;

    // Allow ~299 KB of dynamic LDS (CDNA5 WGP has 320 KB). Idempotent, capture-safe.
    (void)hipFuncSetAttribute((const void*)tree_ensemble_kernel,
                              hipFuncAttributeMaxDynamicSharedMemorySize, (int)SMEM_BYTES);

    dim3 grid(BATCH / TILE_SAMPLES, NUM_TREES / TREES_PER_BLOCK);  // 128 x 32
    tree_ensemble_kernel<<<grid, TILE_SAMPLES, SMEM_BYTES, stream>>>(
        x, lefts, rights, features, thresholds, values, out);
}